// DynamicRoutingLayer_29420525977963
// MI455X (gfx1250) — compile-verified
//
#include <hip/hip_runtime.h>
#include <math.h>

typedef __attribute__((ext_vector_type(2))) float v2f;
typedef __attribute__((ext_vector_type(4))) float f4;
typedef __attribute__((ext_vector_type(8))) float v8f;

#define B_DIM 256
#define N_DIM 512
#define E_DIM 64
#define O_DIM 64
#define K_CAPS 8

// ---------------------------------------------------------------------------
// Kernel 1: priors[r, o] = sum_e emb[r, e] * S[e, o],  r in [0, B*N)
// fp32 WMMA 16x16x4. Block = 256 threads (8 waves) computes a 128x64 slab.
// LDS staged with row stride 68 floats (conflict-free, 16B-aligned rows).
// ---------------------------------------------------------------------------
#define GEMM_TM 128
#define SA_STRIDE 68
#define SB_STRIDE 68

__global__ __launch_bounds__(256) void priors_gemm_wmma(
    const float* __restrict__ emb, const float* __restrict__ S,
    float* __restrict__ priors)
{
  __shared__ float sA[GEMM_TM * SA_STRIDE];   // 34,816 B
  __shared__ float sB[E_DIM * SB_STRIDE];     // 17,408 B

  const int tid  = threadIdx.x;
  const int row0 = blockIdx.x * GEMM_TM;

  // Stage A slab (128 x 64) via float4
  const f4* emb4 = (const f4*)(emb + (size_t)row0 * E_DIM);
  for (int i = tid; i < GEMM_TM * (E_DIM / 4); i += 256) {
    const int m = i >> 4, c4 = i & 15;
    *(f4*)&sA[m * SA_STRIDE + c4 * 4] = emb4[m * 16 + c4];
  }
  // Stage S (64 x 64)
  const f4* S4 = (const f4*)S;
  for (int i = tid; i < E_DIM * (O_DIM / 4); i += 256) {
    const int k = i >> 4, c4 = i & 15;
    *(f4*)&sB[k * SB_STRIDE + c4 * 4] = S4[i];
  }
  __syncthreads();

  const int wave = tid >> 5, lane = tid & 31;
  const int lm = lane & 15;       // row-within-tile / col-within-tile
  const int lh = lane >> 4;       // lane half
  const int m0 = wave * 16;

  for (int nt = 0; nt < O_DIM / 16; ++nt) {
    const int n0 = nt * 16;
    v8f acc;
#pragma unroll
    for (int j = 0; j < 8; ++j) acc[j] = 0.0f;

#pragma unroll
    for (int k0 = 0; k0 < E_DIM; k0 += 4) {
      // A 16x4 f32: lane=M, VGPR j holds K = k0 + 2*lh + j
      v2f a;
      const float* ap = &sA[(m0 + lm) * SA_STRIDE + k0 + 2 * lh];
      a[0] = ap[0];
      a[1] = ap[1];
      // B 4x16 f32: VGPR j holds row K = k0 + j + 2*lh, col N = n0 + lm
      v2f bm;
      bm[0] = sB[(k0 + 0 + 2 * lh) * SB_STRIDE + n0 + lm];
      bm[1] = sB[(k0 + 1 + 2 * lh) * SB_STRIDE + n0 + lm];
      acc = __builtin_amdgcn_wmma_f32_16x16x4_f32(
          /*neg_a=*/false, a, /*neg_b=*/false, bm,
          /*c_mod=*/(short)0, acc, /*reuse_a=*/false, /*reuse_b=*/false);
    }
    // D 16x16 f32: VGPR j -> row m0 + j + 8*lh, col n0 + lm
    float* dst = priors + ((size_t)(row0 + m0 + 8 * lh)) * O_DIM + n0 + lm;
#pragma unroll
    for (int j = 0; j < 8; ++j) dst[(size_t)j * O_DIM] = acc[j];
  }
}

// ---------------------------------------------------------------------------
// Kernel 2: dynamic routing. One workgroup per batch b, 512 threads.
// cc is never written back: cc_t = cc0 + accSim (accSim kept in LDS).
// ---------------------------------------------------------------------------
#define NP 65  // padded priors row stride in LDS (bank = (n + o) % 64)

__global__ __launch_bounds__(512) void routing_kernel(
    const float* __restrict__ priors, const float* __restrict__ cc,
    float* __restrict__ out)
{
  extern __shared__ float smem[];
  float* priorsLds = smem;                              // 512*65   floats
  float* accSim    = priorsLds + N_DIM * NP;            // 8*512
  float* zLds      = accSim + K_CAPS * N_DIM;           // 8*64
  float* vLds      = zLds + K_CAPS * O_DIM;             // 8*64
  float* part      = vLds + K_CAPS * O_DIM;             // 8*8*64
  float* scaleL    = part + K_CAPS * 8 * O_DIM;         // 8

  const int tid = threadIdx.x;
  const int b   = blockIdx.x;
  const float* pb  = priors + (size_t)b * N_DIM * O_DIM;
  const float* ccb = cc + (size_t)b * K_CAPS * N_DIM * O_DIM;

  // Stage priors[b] into padded LDS; zero the sim accumulator.
  for (int i = tid; i < N_DIM * O_DIM; i += 512) {
    const int n = i >> 6, o = i & 63;
    priorsLds[n * NP + o] = pb[i];
  }
  for (int i = tid; i < K_CAPS * N_DIM; i += 512) accSim[i] = 0.0f;
  __syncthreads();

  const int o = tid & 63;   // output feature
  const int g = tid >> 6;   // n-group (0..7); also k for reduce/out phases

  for (int iter = 0; iter < 3; ++iter) {
    // ---- Pass A: w = softmax_K(cc0 + accSim); z[k,o] += w * priors[n,o] ----
    float zacc[K_CAPS];
#pragma unroll
    for (int k = 0; k < K_CAPS; ++k) zacc[k] = 0.0f;

    for (int jn = 0; jn < N_DIM / 8; ++jn) {
      const int n = g + (jn << 3);
      float c[K_CAPS];
      float m = -3.402823466e+38f;
#pragma unroll
      for (int k = 0; k < K_CAPS; ++k) {
        c[k] = ccb[(size_t)k * (N_DIM * O_DIM) + n * O_DIM + o]
             + accSim[k * N_DIM + n];
        m = fmaxf(m, c[k]);
      }
      float s = 0.0f;
#pragma unroll
      for (int k = 0; k < K_CAPS; ++k) { c[k] = __expf(c[k] - m); s += c[k]; }
      const float p = priorsLds[n * NP + o] / s;
#pragma unroll
      for (int k = 0; k < K_CAPS; ++k) zacc[k] += c[k] * p;
    }
#pragma unroll
    for (int k = 0; k < K_CAPS; ++k) part[(k * 8 + g) * O_DIM + o] = zacc[k];
    __syncthreads();

    // ---- Reduce partials over the 8 n-groups -> z[k,o] ----
    {
      float z = 0.0f;
#pragma unroll
      for (int g2 = 0; g2 < 8; ++g2) z += part[(g * 8 + g2) * O_DIM + o];
      zLds[g * O_DIM + o] = z;
    }
    __syncthreads();

    // ---- Squash scale per capsule k ----
    if (tid < K_CAPS) {
      float sq = 0.0f;
      for (int oo = 0; oo < O_DIM; ++oo) {
        const float zz = zLds[tid * O_DIM + oo];
        sq += zz * zz;
      }
      scaleL[tid] = (sq / (1.0f + sq)) / sqrtf(sq + 1e-9f);
    }
    __syncthreads();

    if (iter == 2) {
      // Final: out[b,k,o] = squash(z)
      out[(size_t)b * K_CAPS * O_DIM + tid] = zLds[tid] * scaleL[g];
    } else {
      vLds[tid] = zLds[tid] * scaleL[g];   // v[k,o], tid = k*64 + o
      __syncthreads();
      // ---- Pass B: accSim[k,n] += sum_o priors[n,o] * v[k,o] ----
      const int n = tid;                   // 0..511
#pragma unroll
      for (int k = 0; k < K_CAPS; ++k) {
        float s2 = 0.0f;
#pragma unroll 8
        for (int oo = 0; oo < O_DIM; ++oo)
          s2 += priorsLds[n * NP + oo] * vLds[k * O_DIM + oo];
        accSim[k * N_DIM + n] += s2;
      }
      __syncthreads();
    }
  }
}

// ---------------------------------------------------------------------------
extern "C" void kernel_launch(void* const* d_in, const int* in_sizes, int n_in,
                              void* d_out, int out_size, void* d_ws, size_t ws_size,
                              hipStream_t stream) {
  (void)in_sizes; (void)n_in; (void)out_size; (void)ws_size;
  const float* emb = (const float*)d_in[0];   // (B, N, E)
  const float* S   = (const float*)d_in[1];   // (E, O)
  const float* cc  = (const float*)d_in[2];   // (B, K, N, O)
  float* out    = (float*)d_out;              // (B, K, O)
  float* priors = (float*)d_ws;               // needs B*N*O*4 = 33.5 MB

  priors_gemm_wmma<<<(B_DIM * N_DIM) / GEMM_TM, 256, 0, stream>>>(emb, S, priors);

  const size_t smemBytes =
      (size_t)(N_DIM * NP + K_CAPS * N_DIM + 2 * K_CAPS * O_DIM +
               K_CAPS * 8 * O_DIM + 8) * sizeof(float);   // ~170 KB < 320 KB
  routing_kernel<<<B_DIM, 512, smemBytes, stream>>>(priors, cc, out);
}